// SAGE_11398843203861
// MI455X (gfx1250) — compile-verified
//
#include <hip/hip_runtime.h>

#define N_NODES 50000
#define N_EDGES 800000
#define D 96
#define MTILES (N_NODES / 16)   // 3125, exact
#define NTILES (D / 16)         // 6
#define TILES (MTILES * NTILES) // 18750

typedef __attribute__((ext_vector_type(2))) float v2f;
typedef __attribute__((ext_vector_type(8))) float v8f;

// ---------------------------------------------------------------- zero
__global__ void sage_zero(float* __restrict__ p, int n) {
    int i = blockIdx.x * blockDim.x + threadIdx.x;
    if (i < n) p[i] = 0.0f;
}

// ---------------------------------------------------------------- scatter
// One wave32 per edge: coalesced row read, hardware f32 atomics into agg.
__global__ __launch_bounds__(256) void sage_scatter(
    const float* __restrict__ X, const int* __restrict__ src,
    const int* __restrict__ dst, float* __restrict__ agg,
    float* __restrict__ deg) {
    int wave = (blockIdx.x * blockDim.x + threadIdx.x) >> 5;
    int lane = threadIdx.x & 31;
    if (wave >= N_EDGES) return;
    int u = src[wave];          // wave-uniform -> scalar load
    int v = dst[wave];
    const float* xr = X + (size_t)u * D;
    float* ar = agg + (size_t)v * D;
#pragma unroll
    for (int f = lane; f < D; f += 32)
        unsafeAtomicAdd(&ar[f], xr[f]);       // global_atomic_add_f32
    if (lane == 0) unsafeAtomicAdd(&deg[v], 1.0f);
}

// ---------------------------------------------------------------- fused SAGE layer
// out = [relu]( X*Wself + (AGG/deg)*Wneigh + bias )
// One wave per 16x16 output tile; FP32 WMMA 16x16x4; weights staged
// transposed in LDS ([n][k]) so B fragments are single b64 LDS loads.
__global__ __launch_bounds__(256) void sage_gemm(
    const float* __restrict__ X, const float* __restrict__ AGG,
    const float* __restrict__ DEG, const float* __restrict__ Wself,
    const float* __restrict__ Wneigh, const float* __restrict__ bias,
    float* __restrict__ OUT, int relu) {
    __shared__ float ldsWs[D * D];  // 36 KB, transposed [n][k]
    __shared__ float ldsWn[D * D];  // 36 KB

    for (int i = threadIdx.x; i < D * D; i += 256) {
        int k = i / D, n = i % D;
        ldsWs[n * D + k] = Wself[i];
        ldsWn[n * D + k] = Wneigh[i];
    }
    __syncthreads();

    int wave = blockIdx.x * 8 + (threadIdx.x >> 5);
    if (wave >= TILES) return;                 // whole-wave exit: EXEC stays all-1s
    int lane = threadIdx.x & 31;
    int l16  = lane & 15;
    int half = lane >> 4;

    int mt = wave / NTILES, nt = wave % NTILES;
    int m0 = mt * 16, n0 = nt * 16;

    // A-fragment row for this lane (same row in both halves; halves differ in K)
    int rowA = m0 + l16;
    int colB = n0 + l16;
    float invd = 1.0f / fmaxf(DEG[rowA], 1.0f);

    const float* xrow = X   + (size_t)rowA * D;
    const float* arow = AGG + (size_t)rowA * D;

    v8f c = {};
#pragma unroll
    for (int k0 = 0; k0 < D; k0 += 4) {
        int kb = k0 + 2 * half;                // lanes 0-15: K=k0,k0+1 ; 16-31: K=k0+2,k0+3
        v2f a_self = *(const v2f*)(xrow + kb);               // 8B-aligned global b64
        v2f a_nb   = *(const v2f*)(arow + kb);
        a_nb.x *= invd; a_nb.y *= invd;                      // fold mean into A
        v2f b_self = *(const v2f*)(&ldsWs[colB * D + kb]);   // ds_load_b64
        v2f b_nb   = *(const v2f*)(&ldsWn[colB * D + kb]);
        c = __builtin_amdgcn_wmma_f32_16x16x4_f32(
            false, a_self, false, b_self, (short)0, c, false, false);
        c = __builtin_amdgcn_wmma_f32_16x16x4_f32(
            false, a_nb, false, b_nb, (short)0, c, false, false);
    }

    float bb = bias[colB];
#pragma unroll
    for (int r = 0; r < 8; ++r) {
        int row = m0 + 8 * half + r;           // C/D layout: lanes 16-31 hold M=8..15
        float v = c[r] + bb;
        if (relu) v = fmaxf(v, 0.0f);
        OUT[(size_t)row * D + colB] = v;
    }
}

// ---------------------------------------------------------------- launch
extern "C" void kernel_launch(void* const* d_in, const int* in_sizes, int n_in,
                              void* d_out, int out_size, void* d_ws, size_t ws_size,
                              hipStream_t stream) {
    const float* x   = (const float*)d_in[0];
    const int*   src = (const int*)d_in[1];
    const int*   dst = (const int*)d_in[2];
    const float* Ws1 = (const float*)d_in[3];
    const float* Wn1 = (const float*)d_in[4];
    const float* b1  = (const float*)d_in[5];
    const float* Ws2 = (const float*)d_in[6];
    const float* Wn2 = (const float*)d_in[7];
    const float* b2  = (const float*)d_in[8];
    float* out = (float*)d_out;

    // workspace: agg [N*D] | deg [N] | h [N*D]  (~38.6 MB)
    float* agg = (float*)d_ws;
    float* deg = agg + (size_t)N_NODES * D;
    float* h   = deg + N_NODES;

    const int zn = N_NODES * D + N_NODES;      // agg+deg contiguous
    dim3 blk(256);

    // ----- layer 1
    sage_zero<<<(zn + 255) / 256, blk, 0, stream>>>(agg, zn);
    sage_scatter<<<(N_EDGES + 7) / 8, blk, 0, stream>>>(x, src, dst, agg, deg);
    sage_gemm<<<(TILES + 7) / 8, blk, 0, stream>>>(x, agg, deg, Ws1, Wn1, b1, h, 1);

    // ----- layer 2
    sage_zero<<<(zn + 255) / 256, blk, 0, stream>>>(agg, zn);
    sage_scatter<<<(N_EDGES + 7) / 8, blk, 0, stream>>>(h, src, dst, agg, deg);
    sage_gemm<<<(TILES + 7) / 8, blk, 0, stream>>>(h, agg, deg, Ws2, Wn2, b2, out, 0);
}